// SimpleGATLayer_80513456931510
// MI455X (gfx1250) — compile-verified
//
#include <hip/hip_runtime.h>
#include <hip/hip_bf16.h>
#include <math.h>

// SimpleGAT layer for MI455X (gfx1250, wave32).
//   x:[N,128] f32, W:[128,128] f32, att_src/att_dst:[4,32] f32, edge_index:[2,E] int
// out:[N,128] f32
//
// ws layout (floats): xproj[N*128] | a_src[N*4] | a_dst[N*4] | denom[N*4]

typedef __attribute__((ext_vector_type(2))) float v2f;
typedef __attribute__((ext_vector_type(8))) float v8f;

#define IN_DIM 128
#define HEADS 4
#define OUT_DIM 32
#define HD 128            // HEADS*OUT_DIM
#define LDS_STRIDE 132    // padded row stride (floats): 16B aligned, conflict-free b64 reads

// ---------------------------------------------------------------------------
// Kernel 1: x_proj = x @ W^T via V_WMMA_F32_16X16X4_F32.
// Block = 128 threads = 4 waves; each wave computes a 16x128 slab of x_proj.
// ---------------------------------------------------------------------------
__global__ __launch_bounds__(128) void gat_proj_wmma(
    const float* __restrict__ x, const float* __restrict__ W,
    float* __restrict__ xproj, int N, int mtiles)
{
  __shared__ float ldsW[HD * LDS_STRIDE];

  // Cooperative load of W (row-major [hd][in]) into padded LDS.
  {
    const float4* Wv = (const float4*)W;
    float4* Lv = (float4*)ldsW;
    const int total = HD * IN_DIM / 4;            // 4096 float4
    for (int i = threadIdx.x; i < total; i += blockDim.x) {
      int r = i >> 5;                              // row (32 float4 per row)
      int c4 = i & 31;
      Lv[(r * LDS_STRIDE) / 4 + c4] = Wv[i];       // LDS_STRIDE%4==0 -> 16B aligned
    }
  }
  __syncthreads();

  const int wave = threadIdx.x >> 5;
  const int lane = threadIdx.x & 31;
  const int mtile = blockIdx.x * 4 + wave;
  if (mtile >= mtiles) return;
  const int m0 = mtile * 16;

  const int halfsel = lane >> 4;   // 0: lanes 0-15, 1: lanes 16-31
  const int lrow = lane & 15;
  int arow = m0 + lrow;
  if (arow >= N) arow = N - 1;     // clamp OOB A loads; stores are guarded
  const float* xrow = x + (size_t)arow * IN_DIM;
  const int koff = 2 * halfsel;    // A fp32 16x4 layout: hi half-wave holds K=2,3

  v8f acc[8] = {};

  for (int ks = 0; ks < IN_DIM / 4; ++ks) {
    const int k = 4 * ks + koff;
    v2f a = *(const v2f*)(xrow + k);               // A[M=lrow][k], A[M=lrow][k+1]
#pragma unroll
    for (int nt = 0; nt < 8; ++nt) {
      // B[k][n] = W[n][k]; n = 16*nt + lrow
      v2f b = *(const v2f*)(&ldsW[(16 * nt + lrow) * LDS_STRIDE + k]);
      acc[nt] = __builtin_amdgcn_wmma_f32_16x16x4_f32(
          /*neg_a=*/false, a, /*neg_b=*/false, b,
          /*c_mod=*/(short)0, acc[nt], /*reuse_a=*/false, /*reuse_b=*/false);
    }
  }

  // C/D layout: VGPR i holds (row = i + 8*halfsel, col = lrow) of 16x16 tile.
  const int rbase = 8 * halfsel;
#pragma unroll
  for (int nt = 0; nt < 8; ++nt) {
    const int col = 16 * nt + lrow;
#pragma unroll
    for (int i = 0; i < 8; ++i) {
      const int r = m0 + rbase + i;
      if (r < N) xproj[(size_t)r * HD + col] = acc[nt][i];
    }
  }
}

// ---------------------------------------------------------------------------
// Kernel 2: per-(node,head) attention logits: a = <x_proj[n,h,:], att[h,:]>
// ---------------------------------------------------------------------------
__global__ void gat_node_logits(const float* __restrict__ xproj,
                                const float* __restrict__ att_src,
                                const float* __restrict__ att_dst,
                                float* __restrict__ asrc,
                                float* __restrict__ adst, int N)
{
  int t = blockIdx.x * blockDim.x + threadIdx.x;
  if (t >= N * HEADS) return;
  int n = t >> 2, h = t & 3;
  const float4* xp = (const float4*)(xproj + (size_t)n * HD + h * OUT_DIM);
  const float4* as = (const float4*)(att_src + h * OUT_DIM);
  const float4* ad = (const float4*)(att_dst + h * OUT_DIM);
  float s = 0.f, d = 0.f;
#pragma unroll
  for (int i = 0; i < OUT_DIM / 4; ++i) {
    float4 v = xp[i], a0 = as[i], a1 = ad[i];
    s += v.x * a0.x + v.y * a0.y + v.z * a0.z + v.w * a0.w;
    d += v.x * a1.x + v.y * a1.y + v.z * a1.z + v.w * a1.w;
  }
  asrc[t] = s;
  adst[t] = d;
}

// ---------------------------------------------------------------------------
// Kernel 3: zero fill (d_out and denom are poisoned by the harness)
// ---------------------------------------------------------------------------
__global__ void zero_f32(float* __restrict__ p, size_t n)
{
  size_t i = (size_t)blockIdx.x * blockDim.x + threadIdx.x;
  if (i < n) p[i] = 0.f;
}

__device__ __forceinline__ float lrelu_exp(float t)
{
  t = t > 0.f ? t : 0.2f * t;   // leaky_relu slope 0.2, then exp (no max-sub, as in ref)
  return __expf(t);
}

// Native L2 f32 atomic add, no-return form (TH[0]=0 -> tracked by STOREcnt,
// fire-and-forget RMW resolved at the L2 atomic units). Pinned with inline asm
// so it can never fall back to a CAS loop.
__device__ __forceinline__ void atomic_add_f32(float* p, float v)
{
#if defined(__gfx1250__)
  asm volatile("global_atomic_add_f32 %0, %1, off"
               :
               : "v"(p), "v"(v)
               : "memory");
#else
  __hip_atomic_fetch_add(p, v, __ATOMIC_RELAXED, __HIP_MEMORY_SCOPE_AGENT);
#endif
}

// ---------------------------------------------------------------------------
// Kernel 4: segment-softmax denominator. 1 thread / edge, 4 heads each.
// ---------------------------------------------------------------------------
__global__ void gat_edge_denom(const int* __restrict__ src,
                               const int* __restrict__ dst,
                               const float* __restrict__ asrc,
                               const float* __restrict__ adst,
                               float* __restrict__ denom, int E)
{
  int e = blockIdx.x * blockDim.x + threadIdx.x;
  if (e >= E) return;
  int s = src[e], d = dst[e];
  float4 as = ((const float4*)asrc)[s];
  float4 ad = ((const float4*)adst)[d];
  float* dp = denom + (size_t)d * HEADS;
  atomic_add_f32(dp + 0, lrelu_exp(as.x + ad.x));
  atomic_add_f32(dp + 1, lrelu_exp(as.y + ad.y));
  atomic_add_f32(dp + 2, lrelu_exp(as.z + ad.z));
  atomic_add_f32(dp + 3, lrelu_exp(as.w + ad.w));
}

// ---------------------------------------------------------------------------
// Kernel 5: weighted scatter-aggregation. 1 wave / edge; lane f covers
// features [4f, 4f+3] (head = f>>3). x_proj row read is a coalesced 512B burst
// that stays resident in the 192MB L2; output RMWs are native f32 L2 atomics.
// ---------------------------------------------------------------------------
__global__ __launch_bounds__(256) void gat_edge_aggregate(
    const int* __restrict__ src, const int* __restrict__ dst,
    const float* __restrict__ asrc, const float* __restrict__ adst,
    const float* __restrict__ denom, const float* __restrict__ xproj,
    float* __restrict__ out, int E)
{
  int gid = blockIdx.x * blockDim.x + threadIdx.x;
  int e = gid >> 5;
  if (e >= E) return;
  int lane = gid & 31;
  int s = src[e], d = dst[e];
  int h = lane >> 3;
  float ex = lrelu_exp(asrc[(size_t)s * HEADS + h] + adst[(size_t)d * HEADS + h]);
  float alpha = ex / (denom[(size_t)d * HEADS + h] + 1e-9f);
  float4 xp = ((const float4*)xproj)[(size_t)s * (HD / 4) + lane];
  float* op = out + (size_t)d * HD + lane * 4;
  atomic_add_f32(op + 0, xp.x * alpha);
  atomic_add_f32(op + 1, xp.y * alpha);
  atomic_add_f32(op + 2, xp.z * alpha);
  atomic_add_f32(op + 3, xp.w * alpha);
}

// ---------------------------------------------------------------------------
extern "C" void kernel_launch(void* const* d_in, const int* in_sizes, int n_in,
                              void* d_out, int out_size, void* d_ws, size_t ws_size,
                              hipStream_t stream)
{
  const float* x       = (const float*)d_in[0];
  const float* W       = (const float*)d_in[1];
  const float* att_src = (const float*)d_in[2];
  const float* att_dst = (const float*)d_in[3];
  const int*   edge    = (const int*)d_in[4];   // [2,E]: row0=src, row1=dst

  const int N = in_sizes[0] / IN_DIM;
  const int E = in_sizes[4] / 2;
  const int* src = edge;
  const int* dst = edge + E;

  float* out   = (float*)d_out;
  float* xproj = (float*)d_ws;
  float* asrc  = xproj + (size_t)N * HD;
  float* adst  = asrc + (size_t)N * HEADS;
  float* denom = adst + (size_t)N * HEADS;

  // 1) projection GEMM (WMMA)
  const int mtiles = (N + 15) / 16;
  gat_proj_wmma<<<dim3((mtiles + 3) / 4), dim3(128), 0, stream>>>(x, W, xproj, N, mtiles);

  // 2) per-node logits
  gat_node_logits<<<dim3((N * HEADS + 255) / 256), dim3(256), 0, stream>>>(
      xproj, att_src, att_dst, asrc, adst, N);

  // 3) zero denom and out
  {
    size_t nd = (size_t)N * HEADS;
    zero_f32<<<dim3((unsigned)((nd + 255) / 256)), dim3(256), 0, stream>>>(denom, nd);
    size_t no = (size_t)N * HD;
    zero_f32<<<dim3((unsigned)((no + 255) / 256)), dim3(256), 0, stream>>>(out, no);
  }

  // 4) softmax denominators
  gat_edge_denom<<<dim3((E + 255) / 256), dim3(256), 0, stream>>>(
      src, dst, asrc, adst, denom, E);

  // 5) weighted scatter-aggregation (1 wave per edge)
  gat_edge_aggregate<<<dim3((E + 7) / 8), dim3(256), 0, stream>>>(
      src, dst, asrc, adst, denom, xproj, out, E);
}